// LIFSpike_CW_softsimple_mod_54339926229034
// MI455X (gfx1250) — compile-verified
//
#include <hip/hip_runtime.h>
#include <stdint.h>

// Problem dims (fixed by the reference)
#define BB 32
#define CC 128
#define HH 32
#define WW 32
#define TT 8

#define NPOS (BB * CC * HH * WW)        // 4,194,304 sites
#define BLOCK 256                       // 8 waves (wave32)
#define TILES (NPOS / BLOCK)            // 16,384 tiles
#define NBLOCKS 2048                    // persistent blocks
#define TILES_PER_BLOCK (TILES / NBLOCKS)  // exactly 8 -> fully static unroll
#define TILE_FLOATS (BLOCK * TT)        // 2048 floats = 8 KB per tile
#define TILE_BYTES (TILE_FLOATS * 4)    // 8192 bytes
#define TILE_STRIDE_BYTES ((unsigned)NBLOCKS * (unsigned)TILE_BYTES)  // 16 MB

// ---------------------------------------------------------------------------
// Prologue: precompute all sigmoid(param) values into workspace.
// Layout in P (floats): [0,128)=tau  [128,256)=Vth  [256,384)=leak
//                       [384,512)=reVth  [512,1536)=conduct[t*128+c]
// ---------------------------------------------------------------------------
__device__ __forceinline__ float sigmoidf_dev(float v) {
    return 1.0f / (1.0f + expf(-v));
}

__global__ void lif_params_kernel(const float* __restrict__ tau_p,
                                  const float* __restrict__ Vth_p,
                                  const float* __restrict__ leak_p,
                                  const float* __restrict__ reVth_p,
                                  const float* __restrict__ conduct_p,
                                  float* __restrict__ P) {
    int i = blockIdx.x * blockDim.x + threadIdx.x;
    if (i < CC)                      P[i] = sigmoidf_dev(tau_p[i]);
    else if (i < 2 * CC)             P[i] = sigmoidf_dev(Vth_p[i - CC]);
    else if (i < 3 * CC)             P[i] = sigmoidf_dev(leak_p[i - 2 * CC]);
    else if (i < 4 * CC)             P[i] = sigmoidf_dev(reVth_p[i - 3 * CC]);
    else if (i < 4 * CC + TT * CC)   P[i] = sigmoidf_dev(conduct_p[i - 4 * CC]);
}

// ---------------------------------------------------------------------------
// CDNA5 async data-mover helpers (ASYNCcnt path).
// Each call moves 32 B/lane (two B128s) global -> LDS without touching VGPRs.
// The offset: immediate applies to BOTH the LDS and global addresses.
// ---------------------------------------------------------------------------
__device__ __forceinline__ void issue_tile_async(const float* xbase,
                                                 unsigned goff, unsigned ldsoff) {
#if defined(__HIP_DEVICE_COMPILE__)
    asm volatile(
        "global_load_async_to_lds_b128 %0, %1, %2 offset:0\n\t"
        "global_load_async_to_lds_b128 %0, %1, %2 offset:16"
        :
        : "v"(ldsoff), "v"(goff), "s"(xbase)
        : "memory");
#endif
}

template <int N>
__device__ __forceinline__ void wait_asynccnt() {
#if defined(__HIP_DEVICE_COMPILE__)
    asm volatile("s_wait_asynccnt %0" ::"i"(N) : "memory");
#endif
}

// ---------------------------------------------------------------------------
// Main kernel: each block owns exactly 8 tiles (stride NBLOCKS). Fully
// unrolled double-buffered async-LDS pipeline; every buffer choice, wait
// threshold, and address increment is a compile-time constant.
// ---------------------------------------------------------------------------
__global__ void __launch_bounds__(BLOCK)
lif_main_kernel(const float* __restrict__ x, const float* __restrict__ P,
                float* __restrict__ out) {
    __shared__ __align__(16) float smem[2][TILE_FLOATS];

    const unsigned tid = threadIdx.x;
    // Low 32 bits of a generic pointer into LDS == wave-relative LDS byte addr.
    const unsigned lds0 = (unsigned)(uintptr_t)(&smem[0][tid * TT]);
    const unsigned lds1 = (unsigned)(uintptr_t)(&smem[1][tid * TT]);

    const unsigned base_goff = (unsigned)blockIdx.x * TILE_BYTES + tid * 32u;

    // Prime the pipeline: tile 0 -> buffer 0.
    issue_tile_async(x, base_goff, lds0);

#pragma unroll
    for (int k = 0; k < TILES_PER_BLOCK; ++k) {
        if (k + 1 < TILES_PER_BLOCK) {
            // Prefetch tile k+1 into the other buffer, then retire tile k's
            // two loads (async loads complete in issue order).
            issue_tile_async(x, base_goff + (unsigned)(k + 1) * TILE_STRIDE_BYTES,
                             ((k + 1) & 1) ? lds1 : lds0);
            wait_asynccnt<2>();
        } else {
            wait_asynccnt<0>();
        }

        const unsigned tile = (unsigned)blockIdx.x + (unsigned)k * NBLOCKS;

        // Channel is uniform across the tile -> scalar (s_load) param fetches.
        const unsigned c  = (tile >> 2) & (CC - 1);
        const float tau   = P[c];
        const float Vth   = P[CC + c];
        const float leak  = P[2 * CC + c];
        const float reVth = P[3 * CC + c];

        const float* lp = &smem[k & 1][tid * TT];
        const float4 xa = *(const float4*)(lp);
        const float4 xb = *(const float4*)(lp + 4);
        const float xv[TT] = {xa.x, xa.y, xa.z, xa.w, xb.x, xb.y, xb.z, xb.w};

        float u = 0.0f, o = 0.0f, ov[TT];
#pragma unroll
        for (int t = 0; t < TT; ++t) {
            const float cd = P[4 * CC + t * CC + c];           // scalar load
            const float un = tau * u * (1.0f - o) - leak + xv[t] * cd - reVth * o;
            o = (un - Vth) > 0.0f ? 1.0f : 0.0f;               // Heaviside fwd
            u = un;
            ov[t] = o;
        }

        float* op = out + (size_t)tile * TILE_FLOATS + tid * TT;
        *(float4*)(op)     = make_float4(ov[0], ov[1], ov[2], ov[3]);
        *(float4*)(op + 4) = make_float4(ov[4], ov[5], ov[6], ov[7]);
    }
}

// ---------------------------------------------------------------------------
// Entry point
// ---------------------------------------------------------------------------
extern "C" void kernel_launch(void* const* d_in, const int* in_sizes, int n_in,
                              void* d_out, int out_size, void* d_ws, size_t ws_size,
                              hipStream_t stream) {
    (void)in_sizes; (void)n_in; (void)out_size; (void)ws_size;

    const float* x        = (const float*)d_in[0];
    const float* tau_p    = (const float*)d_in[1];
    const float* Vth_p    = (const float*)d_in[2];
    const float* leak_p   = (const float*)d_in[3];
    const float* reVth_p  = (const float*)d_in[4];
    const float* conduct  = (const float*)d_in[5];
    float* P   = (float*)d_ws;    // 1536 floats = 6 KB of workspace
    float* out = (float*)d_out;

    lif_params_kernel<<<6, 256, 0, stream>>>(tau_p, Vth_p, leak_p, reVth_p,
                                             conduct, P);
    lif_main_kernel<<<NBLOCKS, BLOCK, 0, stream>>>(x, P, out);
}